// GraphAttentionLayer_2791728743044
// MI455X (gfx1250) — compile-verified
//
#include <hip/hip_runtime.h>
#include <math.h>

// GAT layer, fused flash-attention style, fp32 WMMA (V_WMMA_F32_16X16X4_F32).
// B=8, N=2048, Fin=256, Fout=64.

typedef __attribute__((ext_vector_type(2))) float v2f;
typedef __attribute__((ext_vector_type(8))) float v8f;

#define GAT_B     8
#define GAT_N     2048
#define GAT_FIN   256
#define GAT_FOUT  64
#define GAT_ALPHA 0.2f
#define GAT_NEGINF (-9.0e15f)

// ---------------------------------------------------------------------------
// Kernel 1: Wh = h @ W  (per-wave 16x64 tile), plus Wh1 = Wh@a[:64],
// Wh2 = Wh@a[64:], computed in-register from the WMMA accumulators.
// Rows are flattened over (B,N): row = b*N + n.
// ---------------------------------------------------------------------------
__global__ __launch_bounds__(128) void gat_wh_kernel(
    const float* __restrict__ h, const float* __restrict__ W,
    const float* __restrict__ a,
    float* __restrict__ Wh, float* __restrict__ Wh1, float* __restrict__ Wh2)
{
  const int lane = threadIdx.x & 31;
  const int wave = threadIdx.x >> 5;
  const int half = lane >> 4;    // lane half selects K pair (A) / K rows (B)
  const int l16  = lane & 15;
  const int row0 = (blockIdx.x * 4 + wave) * 16;   // 16-row tile base

  v8f c0 = {}, c1 = {}, c2 = {}, c3 = {};
  // A fragment (16x4 f32): lanes 0-15 hold row=l16, K={k,k+1};
  // lanes 16-31 hold row=l16, K={k+2,k+3}.
  const float* __restrict__ hrow = h + (size_t)(row0 + l16) * GAT_FIN + 2 * half;

  for (int k = 0; k < GAT_FIN; k += 4) {
    v2f av; av[0] = hrow[k]; av[1] = hrow[k + 1];
    // B fragment (4x16 f32): lanes 0-15 col=n0+l16 with K rows {k,k+1};
    // lanes 16-31 same cols with K rows {k+2,k+3}.
    const float* wp = W + (size_t)(k + 2 * half) * GAT_FOUT + l16;
    v2f b0, b1, b2, b3;
    b0[0] = wp[0];  b0[1] = wp[GAT_FOUT];
    b1[0] = wp[16]; b1[1] = wp[GAT_FOUT + 16];
    b2[0] = wp[32]; b2[1] = wp[GAT_FOUT + 32];
    b3[0] = wp[48]; b3[1] = wp[GAT_FOUT + 48];
    c0 = __builtin_amdgcn_wmma_f32_16x16x4_f32(false, av, false, b0, (short)0, c0, false, false);
    c1 = __builtin_amdgcn_wmma_f32_16x16x4_f32(false, av, false, b1, (short)0, c1, false, false);
    c2 = __builtin_amdgcn_wmma_f32_16x16x4_f32(false, av, false, b2, (short)0, c2, false, false);
    c3 = __builtin_amdgcn_wmma_f32_16x16x4_f32(false, av, false, b3, (short)0, c3, false, false);
  }

  // C/D layout: VGPR r -> row m = r + 8*half, col = n0 + l16.
  float* outp = Wh + (size_t)row0 * GAT_FOUT;
  #pragma unroll
  for (int r = 0; r < 8; ++r) {
    const int m = r + 8 * half;
    float* orow = outp + (size_t)m * GAT_FOUT + l16;
    orow[0]  = c0[r];
    orow[16] = c1[r];
    orow[32] = c2[r];
    orow[48] = c3[r];
  }

  // Wh1/Wh2: per-lane partial dot over the 4 column tiles, then reduce
  // across the 16 lanes of each half (xor offsets stay within the half).
  const float a10 = a[l16],      a11 = a[16 + l16],
              a12 = a[32 + l16], a13 = a[48 + l16];
  const float a20 = a[64 + l16],  a21 = a[80 + l16],
              a22 = a[96 + l16],  a23 = a[112 + l16];
  float p1[8], p2[8];
  #pragma unroll
  for (int r = 0; r < 8; ++r) {
    p1[r] = c0[r] * a10 + c1[r] * a11 + c2[r] * a12 + c3[r] * a13;
    p2[r] = c0[r] * a20 + c1[r] * a21 + c2[r] * a22 + c3[r] * a23;
  }
  #pragma unroll
  for (int off = 1; off < 16; off <<= 1) {
    #pragma unroll
    for (int r = 0; r < 8; ++r) {
      p1[r] += __shfl_xor(p1[r], off, 32);
      p2[r] += __shfl_xor(p2[r], off, 32);
    }
  }
  if (l16 == 0) {
    #pragma unroll
    for (int r = 0; r < 8; ++r) {
      const int m = r + 8 * half;
      Wh1[row0 + m] = p1[r];
      Wh2[row0 + m] = p2[r];
    }
  }
}

// ---------------------------------------------------------------------------
// Kernel 2: fused attention. Each wave owns a 16-row i-tile; the 4 waves of a
// block share one batch and cooperatively stage each 16x64 Wh_j tile in LDS.
// Online softmax (flash) + WMMA accumulation of P @ Wh_j, then ELU epilogue.
// ---------------------------------------------------------------------------
__global__ __launch_bounds__(128) void gat_attn_kernel(
    const float* __restrict__ Wh, const float* __restrict__ Wh1,
    const float* __restrict__ Wh2, const int* __restrict__ adj,
    float* __restrict__ out)
{
  const int b    = blockIdx.y;
  const int lane = threadIdx.x & 31;
  const int wave = threadIdx.x >> 5;
  const int half = lane >> 4;
  const int l16  = lane & 15;
  const int i0   = (blockIdx.x * 4 + wave) * 16;

  // Padded to stride 72 so the two lane halves (rows jr vs jr+2, +144 dwords
  // = +16 banks mod 64) hit disjoint bank sets.
  __shared__ float lWh[16][72];

  const float* __restrict__ Whb  = Wh  + (size_t)b * GAT_N * GAT_FOUT;
  const float* __restrict__ wh1b = Wh1 + (size_t)b * GAT_N;
  const float* __restrict__ wh2b = Wh2 + (size_t)b * GAT_N;
  const int*   __restrict__ adjrow =
      adj + ((size_t)b * GAT_N + (i0 + l16)) * GAT_N;

  const float wh1i = wh1b[i0 + l16];   // this lane's attention row scalar

  float mrow = -INFINITY;  // running row max (duplicated across halves)
  float lrow = 0.0f;       // running row sum
  v8f acc0 = {}, acc1 = {}, acc2 = {}, acc3 = {};

  const int t  = threadIdx.x;
  const int rr = t >> 3;          // staging: row 0..15
  const int cc = (t & 7) * 8;     // staging: col 0,8,...,56

  for (int j0 = 0; j0 < GAT_N; j0 += 16) {
    __syncthreads();   // previous tile's LDS reads done
    {
      const float* src = Whb + (size_t)(j0 + rr) * GAT_FOUT + cc;
      const float4 v0 = *(const float4*)(src);
      const float4 v1 = *(const float4*)(src + 4);
      float* dst = &lWh[rr][cc];
      dst[0] = v0.x; dst[1] = v0.y; dst[2] = v0.z; dst[3] = v0.w;
      dst[4] = v1.x; dst[5] = v1.y; dst[6] = v1.z; dst[7] = v1.w;
    }
    __syncthreads();

    // Scores for this lane: i = i0+l16, j = j0 + 4k + 2*half + {0,1}
    // (exactly the 16x4 f32 A-fragment layout, chunk k = K-step k).
    float s[8];
    float tmax = -INFINITY;
    #pragma unroll
    for (int k = 0; k < 4; ++k) {
      const int j = j0 + 4 * k + 2 * half;
      const float w2a = wh2b[j], w2c = wh2b[j + 1];
      const int ad0 = adjrow[j], ad1 = adjrow[j + 1];
      float e0 = wh1i + w2a; e0 = e0 > 0.0f ? e0 : GAT_ALPHA * e0;
      float e1 = wh1i + w2c; e1 = e1 > 0.0f ? e1 : GAT_ALPHA * e1;
      e0 = ad0 > 0 ? e0 : GAT_NEGINF;
      e1 = ad1 > 0 ? e1 : GAT_NEGINF;
      s[2 * k]     = e0;
      s[2 * k + 1] = e1;
      tmax = fmaxf(tmax, fmaxf(e0, e1));
    }
    tmax = fmaxf(tmax, __shfl_xor(tmax, 16, 32));
    const float mnew  = fmaxf(mrow, tmax);
    const float scale = __expf(mrow - mnew);   // 0 on first tile (mrow=-inf)
    mrow = mnew;

    float psum = 0.0f;
    #pragma unroll
    for (int k = 0; k < 8; ++k) {
      s[k] = __expf(s[k] - mnew);
      psum += s[k];
    }
    psum += __shfl_xor(psum, 16, 32);
    lrow = lrow * scale + psum;

    // Per-C-row rescale factors: row m's scale lives in lane m (<16).
    float sc[8];
    #pragma unroll
    for (int r = 0; r < 8; ++r) sc[r] = __shfl(scale, r + 8 * half, 32);
    #pragma unroll
    for (int r = 0; r < 8; ++r) {
      acc0[r] *= sc[r]; acc1[r] *= sc[r];
      acc2[r] *= sc[r]; acc3[r] *= sc[r];
    }

    // acc += P(16x16) @ Wh_j(16x64): 4 K-chunks x 4 column tiles.
    #pragma unroll
    for (int k = 0; k < 4; ++k) {
      v2f pa; pa[0] = s[2 * k]; pa[1] = s[2 * k + 1];
      const int jr = 4 * k + 2 * half;   // B-fragment K rows for this half
      v2f b0, b1, b2, b3;
      b0[0] = lWh[jr][l16];      b0[1] = lWh[jr + 1][l16];
      b1[0] = lWh[jr][16 + l16]; b1[1] = lWh[jr + 1][16 + l16];
      b2[0] = lWh[jr][32 + l16]; b2[1] = lWh[jr + 1][32 + l16];
      b3[0] = lWh[jr][48 + l16]; b3[1] = lWh[jr + 1][48 + l16];
      acc0 = __builtin_amdgcn_wmma_f32_16x16x4_f32(false, pa, false, b0, (short)0, acc0, false, false);
      acc1 = __builtin_amdgcn_wmma_f32_16x16x4_f32(false, pa, false, b1, (short)0, acc1, false, false);
      acc2 = __builtin_amdgcn_wmma_f32_16x16x4_f32(false, pa, false, b2, (short)0, acc2, false, false);
      acc3 = __builtin_amdgcn_wmma_f32_16x16x4_f32(false, pa, false, b3, (short)0, acc3, false, false);
    }
  }

  // Epilogue: normalize by row sum, ELU, store.
  const float linv = 1.0f / lrow;
  float li[8];
  #pragma unroll
  for (int r = 0; r < 8; ++r) li[r] = __shfl(linv, r + 8 * half, 32);

  float* outb = out + ((size_t)b * GAT_N + i0) * GAT_FOUT;
  #pragma unroll
  for (int r = 0; r < 8; ++r) {
    const int m = r + 8 * half;
    float* orow = outb + (size_t)m * GAT_FOUT + l16;
    float x;
    x = acc0[r] * li[r]; orow[0]  = x > 0.0f ? x : __expf(x) - 1.0f;
    x = acc1[r] * li[r]; orow[16] = x > 0.0f ? x : __expf(x) - 1.0f;
    x = acc2[r] * li[r]; orow[32] = x > 0.0f ? x : __expf(x) - 1.0f;
    x = acc3[r] * li[r]; orow[48] = x > 0.0f ? x : __expf(x) - 1.0f;
  }
}

// ---------------------------------------------------------------------------
extern "C" void kernel_launch(void* const* d_in, const int* in_sizes, int n_in,
                              void* d_out, int out_size, void* d_ws, size_t ws_size,
                              hipStream_t stream) {
  (void)in_sizes; (void)n_in; (void)out_size; (void)ws_size;
  const float* h   = (const float*)d_in[0];
  const int*   adj = (const int*)d_in[1];
  const float* W   = (const float*)d_in[2];
  const float* a   = (const float*)d_in[3];
  float* out = (float*)d_out;

  // Workspace: Wh [B*N*64] | Wh1 [B*N] | Wh2 [B*N]  (~4.13 MB)
  float* Wh  = (float*)d_ws;
  float* Wh1 = Wh  + (size_t)GAT_B * GAT_N * GAT_FOUT;
  float* Wh2 = Wh1 + (size_t)GAT_B * GAT_N;

  // 4 waves/block, 16 rows/wave => 64 rows/block.
  gat_wh_kernel<<<dim3((GAT_B * GAT_N) / 64), 128, 0, stream>>>(
      h, W, a, Wh, Wh1, Wh2);
  gat_attn_kernel<<<dim3(GAT_N / 64, GAT_B), 128, 0, stream>>>(
      Wh, Wh1, Wh2, adj, out);
}